// SPDBatchNormLie_35519379538486
// MI455X (gfx1250) — compile-verified
//
#include <hip/hip_runtime.h>
#include <math.h>

// ---------------- configuration ----------------
#define LD 68                     // LDS row stride (floats) for 64x64 tiles, pad vs bank conflicts
#define MSZ (64*LD)               // one matrix buffer in LDS (floats)
constexpr int Nm   = 64;
constexpr int TPB  = 256;         // 8 waves (wave32)
constexpr int BATCH = 8192;
constexpr int HEAVY_BLOCKS = 512;
constexpr int MPB  = BATCH / HEAVY_BLOCKS;   // 16 matrices per heavy block
constexpr int MEAN_BLOCKS = 64;
constexpr int MEAN_CHUNK  = BATCH / MEAN_BLOCKS; // 128

constexpr int NS_ITERS_MEAN = 20;  // Newton-Schulz iterations for mean sqrt
constexpr int SQRT_LEVELS   = 3;   // inverse scaling-and-squaring depth for logm
constexpr int NS_ITERS_LOG  = 10;
constexpr int LOG_TAYLOR    = 12;
constexpr int EXP_S         = 5;   // expm scaling levels
constexpr int EXP_M         = 8;   // expm Taylor order

typedef __attribute__((ext_vector_type(2))) float v2f;
typedef __attribute__((ext_vector_type(8))) float v8f;

extern __shared__ float smem[];

// ---------------- collective helpers (blockDim.x == TPB, all end with sync) ----------------

__device__ inline void g2l(float* dst, const float* __restrict__ src) {
  for (int i = threadIdx.x; i < Nm*Nm; i += TPB)
    dst[(i>>6)*LD + (i&63)] = src[i];
  __syncthreads();
}

__device__ inline void l2g(float* __restrict__ dst, const float* src) {
  __syncthreads();
  for (int i = threadIdx.x; i < Nm*Nm; i += TPB)
    dst[i] = src[(i>>6)*LD + (i&63)];
  __syncthreads();
}

__device__ inline void setid(float* M, float s) {
  for (int i = threadIdx.x; i < Nm*Nm; i += TPB) {
    int r = i>>6, c = i&63;
    M[r*LD + c] = (r==c) ? s : 0.0f;
  }
  __syncthreads();
}

__device__ inline void scale_lds(float* M, float s) {
  __syncthreads();
  for (int i = threadIdx.x; i < Nm*Nm; i += TPB) {
    int r = i>>6, c = i&63;
    M[r*LD + c] *= s;
  }
  __syncthreads();
}

__device__ inline void sym_lds(float* M) {
  __syncthreads();
  for (int i = threadIdx.x; i < Nm*Nm; i += TPB) {
    int r = i>>6, c = i&63;
    if (c > r) {
      float a = 0.5f * (M[r*LD + c] + M[c*LD + r]);
      M[r*LD + c] = a;
      M[c*LD + r] = a;
    }
  }
  __syncthreads();
}

__device__ inline void transpose_lds(float* dst, const float* src) {
  __syncthreads();
  for (int i = threadIdx.x; i < Nm*Nm; i += TPB) {
    int r = i>>6, c = i&63;
    dst[c*LD + r] = src[r*LD + c];
  }
  __syncthreads();
}

__device__ inline float fnorm(const float* M) {
  __shared__ float red[TPB];
  __syncthreads();
  float s = 0.0f;
  for (int i = threadIdx.x; i < Nm*Nm; i += TPB) {
    float v = M[(i>>6)*LD + (i&63)];
    s += v*v;
  }
  red[threadIdx.x] = s;
  __syncthreads();
  for (int off = TPB/2; off > 0; off >>= 1) {
    if (threadIdx.x < off) red[threadIdx.x] += red[threadIdx.x + off];
    __syncthreads();
  }
  float r = sqrtf(red[0]);
  __syncthreads();
  return r;
}

// ---------------- 64x64 GEMM via V_WMMA_F32_16X16X4_F32 ----------------
// D = alpha*(A*B) + beta*I.   D must not alias A or B.  A may equal B.
// 8 waves: each wave computes 2 of the 16 output 16x16 tiles.
__device__ inline void gemm64(float* __restrict__ D, const float* A, const float* B,
                              float alpha, float beta)
{
  const int lane = threadIdx.x & 31;
  const int wave = threadIdx.x >> 5;
  const int half = lane >> 4;     // 0: lanes 0-15, 1: lanes 16-31
  const int l16  = lane & 15;
  for (int t = wave; t < 16; t += 8) {
    const int tm = t >> 2, tn = t & 3;
    const int mrow = tm*16 + l16;           // A row for this lane
    const int ncol = tn*16 + l16;           // B/C col for this lane
    v8f acc = {0.f,0.f,0.f,0.f,0.f,0.f,0.f,0.f};
    #pragma unroll
    for (int kb = 0; kb < 64; kb += 4) {
      const int k0 = kb + 2*half;           // A: VGPR0 lanes0-15 K=0, lanes16-31 K=2
      v2f av, bv;
      av.x = A[mrow*LD + k0];
      av.y = A[mrow*LD + k0 + 1];
      bv.x = B[(k0    )*LD + ncol];         // B: row (K) striped across lanes
      bv.y = B[(k0 + 1)*LD + ncol];
      acc = __builtin_amdgcn_wmma_f32_16x16x4_f32(
              /*neg_a=*/false, av, /*neg_b=*/false, bv,
              /*c_mod=*/(short)0, acc, /*reuse_a=*/false, /*reuse_b=*/false);
    }
    #pragma unroll
    for (int v = 0; v < 8; ++v) {
      const int row = tm*16 + half*8 + v;   // C/D: VGPR v -> M=v (lo half) / M=v+8 (hi half)
      D[row*LD + ncol] = alpha*acc[v] + ((row == ncol) ? beta : 0.0f);
    }
  }
  __syncthreads();
}

// ---------------- coupled Newton-Schulz matrix sqrt / inv-sqrt ----------------
// Consumes p0 (input A). Uses p1,p2,p3. Outputs pointers to sqrt(A) and A^{-1/2}.
__device__ inline void ns_sqrt(float* p0, float* p1, float* p2, float* p3, int iters,
                               float** sqrtOut, float** invOut)
{
  float c = fnorm(p0);
  scale_lds(p0, 1.0f / c);      // Y0 = A/c
  setid(p1, 1.0f);              // Z0 = I
  float *Y = p0, *Z = p1, *W = p2, *T = p3;
  for (int k = 0; k < iters; ++k) {
    gemm64(T, Z, Y, -0.5f, 1.5f);   // T = 1.5 I - 0.5 Z*Y
    gemm64(W, Y, T,  1.0f, 0.0f);   // Ynew
    gemm64(Y, T, Z,  1.0f, 0.0f);   // Znew (into old-Y buffer; old Y dead)
    float* tmp = Y; Y = W; W = Z; Z = tmp;
  }
  float sc = sqrtf(c);
  __syncthreads();
  for (int i = threadIdx.x; i < Nm*Nm; i += TPB) {
    int r = i>>6, cc = i&63;
    Y[r*LD + cc] *= sc;           // sqrt(A)  = sqrt(c) * Y
    Z[r*LD + cc] *= 1.0f / sc;    // A^{-1/2} = Z / sqrt(c)
  }
  __syncthreads();
  *sqrtOut = Y;
  *invOut  = Z;
}

// ---------------- logm via inverse scaling-and-squaring + Taylor ----------------
// Input X in b0 (destroyed). Returns pointer (within {b0..b3}) to log(X).
__device__ inline float* logm64(float* b0, float* b1, float* b2, float* b3)
{
  float* R  = b0;
  float* o1 = b1; float* o2 = b2; float* o3 = b3;
  for (int l = 0; l < SQRT_LEVELS; ++l) {
    float *sq, *iv;
    ns_sqrt(R, o1, o2, o3, NS_ITERS_LOG, &sq, &iv);
    float* all[4] = {R, o1, o2, o3};
    float* rest[3]; int n = 0;
    for (int i = 0; i < 4; ++i) if (all[i] != sq) rest[n++] = all[i];
    R = sq; o1 = rest[0]; o2 = rest[1]; o3 = rest[2];
  }
  // E = R - I (in place)
  __syncthreads();
  for (int i = threadIdx.x; i < Nm; i += TPB) R[i*LD + i] -= 1.0f;
  __syncthreads();
  // log(I+E) = E * (I - E/2 + E^2/3 - ...)  via Horner
  float* H = o1; float* W = o2;
  float cm = ((LOG_TAYLOR & 1) ? 1.0f : -1.0f) / (float)LOG_TAYLOR;
  setid(H, cm);
  for (int k = LOG_TAYLOR - 1; k >= 1; --k) {
    float ck = ((k & 1) ? 1.0f : -1.0f) / (float)k;
    gemm64(W, R, H, 1.0f, ck);
    float* t = H; H = W; W = t;
  }
  gemm64(W, R, H, (float)(1 << SQRT_LEVELS), 0.0f);
  return W;
}

// ---------------- expm via scaling-and-squaring ----------------
// Input E in b0 (destroyed), overall pre-multiplier `pre`. Returns ptr in {b1,b2}.
__device__ inline float* expm64(float* E, float* h1, float* h2, float pre)
{
  scale_lds(E, pre / (float)(1 << EXP_S));
  float* H = h1; float* W = h2;
  setid(H, 1.0f);
  for (int k = EXP_M; k >= 1; --k) {
    gemm64(W, E, H, 1.0f / (float)k, 1.0f);   // H = I + (E*H)/k
    float* t = H; H = W; W = t;
  }
  for (int s = 0; s < EXP_S; ++s) {
    gemm64(W, H, H, 1.0f, 0.0f);
    float* t = H; H = W; W = t;
  }
  return H;
}

// ---------------- Cholesky (A -> L, lower) and lower-triangular inverse ----------------
__device__ inline void cholesky64(const float* A, float* L)
{
  __syncthreads();
  for (int i = threadIdx.x; i < Nm*Nm; i += TPB) {
    int r = i>>6, c = i&63;
    L[r*LD + c] = (c <= r) ? A[r*LD + c] : 0.0f;
  }
  __syncthreads();
  for (int j = 0; j < Nm; ++j) {
    if (threadIdx.x == 0) {
      float s = L[j*LD + j];
      for (int k = 0; k < j; ++k) { float v = L[j*LD + k]; s -= v*v; }
      L[j*LD + j] = sqrtf(fmaxf(s, 1e-20f));
    }
    __syncthreads();
    float dj = L[j*LD + j];
    for (int i = j + 1 + (int)threadIdx.x; i < Nm; i += TPB) {
      float s = L[i*LD + j];
      for (int k = 0; k < j; ++k) s -= L[i*LD + k] * L[j*LD + k];
      L[i*LD + j] = s / dj;
    }
    __syncthreads();
  }
}

__device__ inline void trinv64(const float* L, float* M)
{
  __syncthreads();
  for (int i = threadIdx.x; i < Nm*Nm; i += TPB)
    M[(i>>6)*LD + (i&63)] = 0.0f;
  __syncthreads();
  if (threadIdx.x < (unsigned)Nm) {
    int c = threadIdx.x;                 // one column per thread, forward substitution
    for (int i = c; i < Nm; ++i) {
      float s = (i == c) ? 1.0f : 0.0f;
      for (int k = c; k < i; ++k) s -= L[i*LD + k] * M[k*LD + c];
      M[i*LD + c] = s / L[i*LD + i];
    }
  }
  __syncthreads();
}

// ================= kernels =================

__global__ void __launch_bounds__(TPB) k_mean1(const float* __restrict__ src,
                                               float* __restrict__ part)
{
  const size_t p = blockIdx.x;
  const float* base = src + p * (size_t)MEAN_CHUNK * 4096;
  for (int e = threadIdx.x; e < 4096; e += TPB) {
    float s = 0.0f;
    for (int m = 0; m < MEAN_CHUNK; ++m) s += base[(size_t)m * 4096 + e];
    part[p * 4096 + e] = s;
  }
}

__global__ void __launch_bounds__(TPB) k_mean2(const float* __restrict__ part,
                                               float* __restrict__ dst, float scale)
{
  for (int e = threadIdx.x; e < 4096; e += TPB) {
    float s = 0.0f;
    for (int p = 0; p < MEAN_BLOCKS; ++p) s += part[(size_t)p * 4096 + e];
    dst[e] = s * scale;
  }
}

__global__ void __launch_bounds__(TPB) k_sqrtmean(const float* __restrict__ G,
                                                  float* __restrict__ Gh,
                                                  float* __restrict__ Gih)
{
  float* b0 = smem; float* b1 = smem + MSZ; float* b2 = smem + 2*MSZ; float* b3 = smem + 3*MSZ;
  g2l(b0, G);
  float *sq, *iv;
  ns_sqrt(b0, b1, b2, b3, NS_ITERS_MEAN, &sq, &iv);
  l2g(Gh,  sq);
  l2g(Gih, iv);
}

__global__ void __launch_bounds__(TPB) k_whiten_log(const float* __restrict__ X,
                                                    const float* __restrict__ Gih,
                                                    float* __restrict__ out)
{
  float* P  = smem;
  float* b0 = smem + MSZ; float* b1 = smem + 2*MSZ;
  float* b2 = smem + 3*MSZ; float* b3 = smem + 4*MSZ;
  g2l(P, Gih);
  for (int m = 0; m < MPB; ++m) {
    const size_t b = (size_t)blockIdx.x * MPB + m;
    g2l(b0, X + b * 4096);
    if (m + 1 < MPB)
      __builtin_prefetch(X + (b + 1) * 4096 + (size_t)threadIdx.x * 16, 0, 1);
    gemm64(b1, P, b0, 1.0f, 0.0f);     // t  = Gih * X
    gemm64(b0, b1, P, 1.0f, 0.0f);     // Xw = t * Gih
    sym_lds(b0);
    float* L = logm64(b0, b1, b2, b3);
    l2g(out + b * 4096, L);
  }
}

__global__ void __launch_bounds__(TPB) k_update(const float* __restrict__ tang,
                                                const float* __restrict__ Gh,
                                                const float* __restrict__ bias,
                                                float* __restrict__ Linv,
                                                float* __restrict__ LinvT,
                                                float* __restrict__ Lb,
                                                float* __restrict__ LbT)
{
  float* b0 = smem; float* b1 = smem + MSZ; float* b2 = smem + 2*MSZ; float* b3 = smem + 3*MSZ;
  g2l(b0, tang);
  float* E  = expm64(b0, b1, b2, 1.0f);          // expm(tang) in b1 or b2
  float* f1 = (E == b1) ? b2 : b1;
  g2l(b0, Gh);                                    // b0 = Gh
  gemm64(f1, b0, E, 1.0f, 0.0f);                  // t  = Gh * expm(tang)
  gemm64(b3, f1, b0, 1.0f, 0.0f);                 // Gn = t * Gh   (Gh symmetric)
  sym_lds(b3);
  cholesky64(b3, f1);                             // L = chol(Gn)
  trinv64(f1, E);                                 // E buffer := L^{-1}
  l2g(Linv, E);
  transpose_lds(b0, E);
  l2g(LinvT, b0);
  // bias Cholesky
  g2l(b3, bias);
  cholesky64(b3, b0);                             // Lb
  l2g(Lb, b0);
  transpose_lds(E, b0);
  l2g(LbT, E);
}

__global__ void __launch_bounds__(TPB) k_center_log(const float* __restrict__ X,
                                                    const float* __restrict__ Linv,
                                                    const float* __restrict__ LinvT,
                                                    float* __restrict__ out,
                                                    float* __restrict__ varPart)
{
  float* P1 = smem;           float* P2 = smem + MSZ;
  float* b0 = smem + 2*MSZ;   float* b1 = smem + 3*MSZ;
  float* b2 = smem + 4*MSZ;   float* b3 = smem + 5*MSZ;
  __shared__ float red[TPB];
  g2l(P1, Linv);
  g2l(P2, LinvT);
  float vacc = 0.0f;
  for (int m = 0; m < MPB; ++m) {
    const size_t b = (size_t)blockIdx.x * MPB + m;
    g2l(b0, X + b * 4096);
    if (m + 1 < MPB)
      __builtin_prefetch(X + (b + 1) * 4096 + (size_t)threadIdx.x * 16, 0, 1);
    gemm64(b1, P1, b0, 1.0f, 0.0f);    // t  = Linv * X
    gemm64(b0, b1, P2, 1.0f, 0.0f);    // Xc = t * Linv^T
    sym_lds(b0);
    float* L = logm64(b0, b1, b2, b3);
    __syncthreads();
    for (int i = threadIdx.x; i < Nm*Nm; i += TPB) {
      float v = L[(i>>6)*LD + (i&63)];
      out[b * 4096 + i] = v;
      vacc += v * v;
    }
    __syncthreads();
  }
  red[threadIdx.x] = vacc;
  __syncthreads();
  for (int off = TPB/2; off > 0; off >>= 1) {
    if (threadIdx.x < off) red[threadIdx.x] += red[threadIdx.x + off];
    __syncthreads();
  }
  if (threadIdx.x == 0) varPart[blockIdx.x] = red[0];
}

__global__ void __launch_bounds__(TPB) k_final(const float* __restrict__ Lc,
                                               const float* __restrict__ Lb,
                                               const float* __restrict__ LbT,
                                               const float* __restrict__ varPart,
                                               const float* __restrict__ shiftp,
                                               float* __restrict__ out)
{
  float* P1 = smem;           float* P2 = smem + MSZ;
  float* b0 = smem + 2*MSZ;   float* b1 = smem + 3*MSZ;  float* b2 = smem + 4*MSZ;
  __shared__ float red[TPB];
  __shared__ float s_factor;
  g2l(P1, Lb);
  g2l(P2, LbT);
  // factor = shift / sqrt(var + EPS); var = sum(varPart)/BATCH (deterministic per-block)
  float s = 0.0f;
  for (int p = threadIdx.x; p < HEAVY_BLOCKS; p += TPB) s += varPart[p];
  red[threadIdx.x] = s;
  __syncthreads();
  for (int off = TPB/2; off > 0; off >>= 1) {
    if (threadIdx.x < off) red[threadIdx.x] += red[threadIdx.x + off];
    __syncthreads();
  }
  if (threadIdx.x == 0) {
    float var = red[0] / (float)BATCH;
    s_factor = shiftp[0] / sqrtf(var + 1e-5f);
  }
  __syncthreads();
  const float factor = s_factor;
  for (int m = 0; m < MPB; ++m) {
    const size_t b = (size_t)blockIdx.x * MPB + m;
    g2l(b0, Lc + b * 4096);
    float* S  = expm64(b0, b1, b2, factor);   // Xs = expm(factor * Lc)
    float* f1 = (S == b1) ? b2 : b1;
    gemm64(f1, P1, S, 1.0f, 0.0f);            // t   = Lb * Xs
    gemm64(b0, f1, P2, 1.0f, 0.0f);           // out = t * Lb^T
    sym_lds(b0);
    l2g(out + b * 4096, b0);
  }
}

// ================= launcher =================

extern "C" void kernel_launch(void* const* d_in, const int* in_sizes, int n_in,
                              void* d_out, int out_size, void* d_ws, size_t ws_size,
                              hipStream_t stream)
{
  (void)in_sizes; (void)n_in; (void)out_size; (void)ws_size;
  const float* X     = (const float*)d_in[0];   // (8192,64,64) f32
  const float* bias  = (const float*)d_in[1];   // (1,64,64)    f32
  const float* shift = (const float*)d_in[2];   // scalar       f32
  float* out = (float*)d_out;
  float* ws  = (float*)d_ws;

  float* mp    = ws;                     // 64*4096 mean/tangent partials
  float* G     = mp    + (size_t)MEAN_BLOCKS * 4096;
  float* Gh    = G     + 4096;
  float* Gih   = Gh    + 4096;
  float* tang  = Gih   + 4096;
  float* Linv  = tang  + 4096;
  float* LinvT = Linv  + 4096;
  float* Lbw   = LinvT + 4096;
  float* LbTw  = Lbw   + 4096;
  float* varP  = LbTw  + 4096;           // HEAVY_BLOCKS floats

  const size_t MB = (size_t)MSZ * sizeof(float);

  // 1) batch mean G (two-stage, deterministic)
  k_mean1<<<MEAN_BLOCKS, TPB, 0, stream>>>(X, mp);
  k_mean2<<<1, TPB, 0, stream>>>(mp, G, 1.0f / (float)BATCH);
  // 2) G^{1/2}, G^{-1/2} via Newton-Schulz
  k_sqrtmean<<<1, TPB, 4 * MB, stream>>>(G, Gh, Gih);
  // 3) per-batch whitening + logm -> d_out (scratch)
  k_whiten_log<<<HEAVY_BLOCKS, TPB, 5 * MB, stream>>>(X, Gih, out);
  // 4) tangent mean from d_out
  k_mean1<<<MEAN_BLOCKS, TPB, 0, stream>>>(out, mp);
  k_mean2<<<1, TPB, 0, stream>>>(mp, tang, 1.0f / (float)BATCH);
  // 5) Karcher update, Cholesky factors and inverses
  k_update<<<1, TPB, 4 * MB, stream>>>(tang, Gh, bias, Linv, LinvT, Lbw, LbTw);
  // 6) centering + logm -> d_out, variance partials
  k_center_log<<<HEAVY_BLOCKS, TPB, 6 * MB, stream>>>(X, Linv, LinvT, out, varP);
  // 7) scale on log domain, expm, bias translation -> d_out
  k_final<<<HEAVY_BLOCKS, TPB, 5 * MB, stream>>>(out, Lbw, LbTw, varP, shift, out);
}